// Encoder_24532853195515
// MI455X (gfx1250) — compile-verified
//
#include <hip/hip_runtime.h>
#include <hip/hip_bf16.h>

// ---------------------------------------------------------------------------
// PointNet++ encoder for MI455X (gfx1250, wave32).
// Matrix work uses V_WMMA_F32_16X16X4_F32 (full f32 precision matching the
// reference). Channel dims padded to multiples of 4 so the WMMA K-loop is
// branch-free (no EXEC manipulation). All scratch in d_ws, all on `stream`.
// ---------------------------------------------------------------------------

typedef float v2f __attribute__((ext_vector_type(2)));
typedef float v8f __attribute__((ext_vector_type(8)));

static constexpr int BB = 32;    // batch
static constexpr int NN = 2048;  // points

// ---------------------------------------------------------------------------
// split input (B,N,6) -> xyz0 (B,N,3) row-major  and feat0 (B,6,N) chan-major
// ---------------------------------------------------------------------------
__global__ __launch_bounds__(256) void split_input(const float* __restrict__ in,
                                                   float* __restrict__ xyz0,
                                                   float* __restrict__ feat0) {
  size_t gid = (size_t)blockIdx.x * 256 + threadIdx.x;
  size_t tot = (size_t)BB * NN;
  if (gid >= tot) return;
  size_t b = gid / NN, n = gid % NN;
  const float* p = in + gid * 6;
  for (int c = 0; c < 6; ++c) {
    float v = p[c];
    if (c < 3) xyz0[gid * 3 + c] = v;
    feat0[((size_t)b * 6 + c) * NN + n] = v;
  }
}

// ---------------------------------------------------------------------------
// Farthest point sampling: one block per batch, 256 threads, iterative argmax.
// Mirrors reference: out[0]=0, dist init 1e10, min-update, argmax (lowest idx
// wins ties, matching jnp.argmax).
// ---------------------------------------------------------------------------
__global__ __launch_bounds__(256) void fps_kernel(const float* __restrict__ xyz,
                                                  int* __restrict__ out,
                                                  int Np, int npoint) {
  const int b = blockIdx.x;
  const int t = threadIdx.x;
  const float* xb = xyz + (size_t)b * Np * 3;
  float dloc[8];
  const int per = (Np + 255) >> 8;
  for (int i = 0; i < per; ++i) dloc[i] = 1e10f;
  __shared__ float sv[256];
  __shared__ int   si[256];
  __shared__ float cp[3];
  int far = 0;
  for (int it = 0; it < npoint; ++it) {
    if (t == 0) {
      out[(size_t)b * npoint + it] = far;
      cp[0] = xb[(size_t)far * 3];
      cp[1] = xb[(size_t)far * 3 + 1];
      cp[2] = xb[(size_t)far * 3 + 2];
    }
    __syncthreads();
    float cx = cp[0], cy = cp[1], cz = cp[2];
    float best = -1.f; int bi = 0;
    for (int i = 0; i < per; ++i) {
      int n = i * 256 + t;
      if (n < Np) {
        float dx = xb[(size_t)n * 3]     - cx;
        float dy = xb[(size_t)n * 3 + 1] - cy;
        float dz = xb[(size_t)n * 3 + 2] - cz;
        float d  = dx * dx + dy * dy + dz * dz;
        float nd = fminf(dloc[i], d);
        dloc[i] = nd;
        if (nd > best) { best = nd; bi = n; }
      }
    }
    sv[t] = best; si[t] = bi;
    __syncthreads();
    for (int s = 128; s > 0; s >>= 1) {
      if (t < s) {
        if (sv[t + s] > sv[t] || (sv[t + s] == sv[t] && si[t + s] < si[t])) {
          sv[t] = sv[t + s]; si[t] = si[t + s];
        }
      }
      __syncthreads();
    }
    far = si[0];
    __syncthreads();
  }
}

// gather sampled coordinates: new_xyz[b,s,:] = xyz[b, fidx[b,s], :]
__global__ __launch_bounds__(256) void gather_new(const float* __restrict__ xyz,
                                                  const int* __restrict__ fidx,
                                                  float* __restrict__ nxyz,
                                                  int Np, int S) {
  size_t gid = (size_t)blockIdx.x * 256 + threadIdx.x;
  size_t tot = (size_t)BB * S;
  if (gid >= tot) return;
  size_t b = gid / S;
  int f = fidx[gid];
  for (int c = 0; c < 3; ++c)
    nxyz[gid * 3 + c] = xyz[((size_t)b * Np + f) * 3 + c];
}

// ---------------------------------------------------------------------------
// ball query: first K indices (ascending) with sqdist < r2, pad with first.
// ---------------------------------------------------------------------------
__global__ __launch_bounds__(256) void query_ball(const float* __restrict__ xyz,
                                                  const float* __restrict__ nxyz,
                                                  int* __restrict__ idx,
                                                  int Np, int S, int K, float r2) {
  size_t gid = (size_t)blockIdx.x * 256 + threadIdx.x;
  size_t tot = (size_t)BB * S;
  if (gid >= tot) return;
  size_t b = gid / S;
  const float* q = nxyz + gid * 3;
  float qx = q[0], qy = q[1], qz = q[2];
  const float* xb = xyz + (size_t)b * Np * 3;
  int* ob = idx + gid * K;
  int cnt = 0, first = -1;
  for (int n = 0; n < Np && cnt < K; ++n) {
    float dx = xb[(size_t)n * 3] - qx;
    float dy = xb[(size_t)n * 3 + 1] - qy;
    float dz = xb[(size_t)n * 3 + 2] - qz;
    float d = dx * dx + dy * dy + dz * dz;
    if (d < r2) { if (first < 0) first = n; ob[cnt++] = n; }
  }
  int p = (first >= 0) ? first : 0;
  for (; cnt < K; ++cnt) ob[cnt] = p;
}

// ---------------------------------------------------------------------------
// build grouped GEMM input [B, Cpad, S*K]: centered xyz, gathered feats, then
// zero-fill pad channels (Cpad = round-up-4 of 3+Cp).
// pts is [B, Ctot, Np], features taken from channel offset coff.
// ---------------------------------------------------------------------------
__global__ __launch_bounds__(256) void group_kernel(const float* __restrict__ xyz,
                                                    const float* __restrict__ nxyz,
                                                    const float* __restrict__ pts,
                                                    const int* __restrict__ idx,
                                                    float* __restrict__ out,
                                                    int Np, int S, int K,
                                                    int Ctot, int coff, int Cp,
                                                    int Cpad) {
  size_t gid = (size_t)blockIdx.x * 256 + threadIdx.x;
  size_t tot = (size_t)BB * S * K;
  if (gid >= tot) return;
  size_t SK = (size_t)S * K;
  size_t b = gid / SK, rem = gid % SK;
  size_t s = rem / K;
  int id = idx[gid];
  float* ob = out + (size_t)b * Cpad * SK + rem;
  for (int c = 0; c < 3; ++c)
    ob[(size_t)c * SK] = xyz[((size_t)b * Np + id) * 3 + c] -
                         nxyz[((size_t)b * S + s) * 3 + c];
  for (int c = 0; c < Cp; ++c)
    ob[(size_t)(3 + c) * SK] = pts[((size_t)b * Ctot + coff + c) * Np + id];
  for (int c = 3 + Cp; c < Cpad; ++c)
    ob[(size_t)c * SK] = 0.f;
}

// sa3 input: [B, 260, 128] = concat(l2_xyz(3), l2_pts(256), zero-pad(1))
__global__ __launch_bounds__(256) void build_sa3(const float* __restrict__ l2xyz,
                                                 const float* __restrict__ l2pts,
                                                 float* __restrict__ out) {
  size_t gid = (size_t)blockIdx.x * 256 + threadIdx.x;
  size_t tot = (size_t)BB * 260 * 128;
  if (gid >= tot) return;
  size_t b = gid / (260 * 128);
  size_t rem = gid % (260 * 128);
  int c = (int)(rem / 128), m = (int)(rem % 128);
  float v;
  if (c < 3)        v = l2xyz[((size_t)b * 128 + m) * 3 + c];
  else if (c < 259) v = l2pts[((size_t)b * 256 + (c - 3)) * 128 + m];
  else              v = 0.f;
  out[gid] = v;
}

// ---------------------------------------------------------------------------
// 3-NN with inverse-distance weights (stable insertion = stable top_k ties)
// ---------------------------------------------------------------------------
__global__ __launch_bounds__(256) void knn3_kernel(const float* __restrict__ xyz1,
                                                   const float* __restrict__ xyz2,
                                                   int* __restrict__ kidx,
                                                   float* __restrict__ kw,
                                                   int N1, int N2) {
  size_t gid = (size_t)blockIdx.x * 256 + threadIdx.x;
  size_t tot = (size_t)BB * N1;
  if (gid >= tot) return;
  size_t b = gid / N1;
  const float* q = xyz1 + gid * 3;
  float qx = q[0], qy = q[1], qz = q[2];
  const float* xb = xyz2 + (size_t)b * N2 * 3;
  float d0 = 3.4e38f, d1 = 3.4e38f, d2 = 3.4e38f;
  int i0 = 0, i1 = 0, i2 = 0;
  for (int j = 0; j < N2; ++j) {
    float dx = xb[(size_t)j * 3] - qx;
    float dy = xb[(size_t)j * 3 + 1] - qy;
    float dz = xb[(size_t)j * 3 + 2] - qz;
    float d = dx * dx + dy * dy + dz * dz;
    if (d < d0)      { d2 = d1; i2 = i1; d1 = d0; i1 = i0; d0 = d; i0 = j; }
    else if (d < d1) { d2 = d1; i2 = i1; d1 = d; i1 = j; }
    else if (d < d2) { d2 = d; i2 = j; }
  }
  d0 = fmaxf(d0, 1e-10f); d1 = fmaxf(d1, 1e-10f); d2 = fmaxf(d2, 1e-10f);
  float w0 = 1.f / d0, w1 = 1.f / d1, w2 = 1.f / d2;
  float ws = w0 + w1 + w2;
  kidx[gid * 3] = i0; kidx[gid * 3 + 1] = i1; kidx[gid * 3 + 2] = i2;
  kw[gid * 3] = w0 / ws; kw[gid * 3 + 1] = w1 / ws; kw[gid * 3 + 2] = w2 / ws;
}

// fp input: [B, Cpad, N1]; mode0 = broadcast pts2 (N2==1), mode1 = 3-NN interp,
// zero-fill channels in [C2+C1, Cpad)
__global__ __launch_bounds__(256) void build_fp_input(const float* __restrict__ pts2,
                                                      const float* __restrict__ pts1,
                                                      const int* __restrict__ kidx,
                                                      const float* __restrict__ kw,
                                                      float* __restrict__ out,
                                                      int C2, int N2, int C1, int N1,
                                                      int Cpad, int mode) {
  size_t gid = (size_t)blockIdx.x * 256 + threadIdx.x;
  size_t tot = (size_t)BB * Cpad * N1;
  if (gid >= tot) return;
  size_t b = gid / ((size_t)Cpad * N1);
  size_t rem = gid % ((size_t)Cpad * N1);
  int c = (int)(rem / N1), n = (int)(rem % N1);
  float v;
  if (c < C2) {
    if (mode == 0) {
      v = pts2[(size_t)b * C2 + c];
    } else {
      const int*   ki = kidx + ((size_t)b * N1 + n) * 3;
      const float* wv = kw   + ((size_t)b * N1 + n) * 3;
      const float* p  = pts2 + ((size_t)b * C2 + c) * N2;
      v = wv[0] * p[ki[0]] + wv[1] * p[ki[1]] + wv[2] * p[ki[2]];
    }
  } else if (c < C2 + C1) {
    v = pts1[((size_t)b * C1 + (c - C2)) * N1 + n];
  } else {
    v = 0.f;
  }
  out[gid] = v;
}

// zero-pad weight copy: Wp[o, 0..Cpad) = W[o, 0..C) then zeros
__global__ __launch_bounds__(256) void pad_w(const float* __restrict__ W,
                                             float* __restrict__ Wp,
                                             int C, int Cpad, int O) {
  int i = blockIdx.x * 256 + threadIdx.x;
  if (i >= O * Cpad) return;
  int o = i / Cpad, c = i % Cpad;
  Wp[i] = (c < C) ? W[(size_t)o * C + c] : 0.f;
}

// ---------------------------------------------------------------------------
// GEMM via V_WMMA_F32_16X16X4_F32: Y[b,o,m] = sum_c W[o,c] X[b,c,m] + bias[o]
// Wave computes 16(o) x 32(m) with two accumulators sharing one A fragment.
// Block = 8 waves -> 32(o) x 128(m). Cpad % 4 == 0 -> branch-free K-loop,
// EXEC all-ones at every WMMA. BN sum/sumsq via in-wave reduce + atomics.
// ---------------------------------------------------------------------------
__global__ __launch_bounds__(256) void gemm_bn_stats(const float* __restrict__ X,
                                                     const float* __restrict__ W,
                                                     const float* __restrict__ bias,
                                                     float* __restrict__ Y,
                                                     float* __restrict__ stats,
                                                     int Cpad, int O, int M) {
  const int b    = blockIdx.z;
  const int lane = threadIdx.x & 31;
  const int wave = threadIdx.x >> 5;
  const int o_base = blockIdx.y * 32 + (wave >> 2) * 16;
  const int m_base = blockIdx.x * 128 + (wave & 3) * 32;
  const int row  = lane & 15;
  const int koff = (lane >> 4) << 1;   // lanes 0-15: K=0,1 ; lanes 16-31: K=2,3

  const float* Wr = W + (size_t)(o_base + row) * Cpad + koff;
  const float* Xr = X + (size_t)b * Cpad * M + m_base + row;

  v8f acc0 = {0.f, 0.f, 0.f, 0.f, 0.f, 0.f, 0.f, 0.f};
  v8f acc1 = {0.f, 0.f, 0.f, 0.f, 0.f, 0.f, 0.f, 0.f};

  for (int c0 = 0; c0 < Cpad; c0 += 4) {
    v2f a;                         // contiguous pair -> single b64 load
    a.x = Wr[c0];
    a.y = Wr[c0 + 1];
    const float* xc = Xr + (size_t)(c0 + koff) * M;
    v2f b0, b1;
    b0.x = xc[0];        b0.y = xc[M];
    b1.x = xc[16];       b1.y = xc[M + 16];
    __builtin_prefetch(xc + 4 * (size_t)M, 0, 1);   // speculative, branch-free
    acc0 = __builtin_amdgcn_wmma_f32_16x16x4_f32(
        false, a, false, b0, (short)0, acc0, false, false);
    acc1 = __builtin_amdgcn_wmma_f32_16x16x4_f32(
        false, a, false, b1, (short)0, acc1, false, false);
  }

  const int half = lane >> 4;
  for (int r = 0; r < 8; ++r) {
    int o = o_base + r + (half << 3);
    float bo = bias[o];
    float v0 = acc0[r] + bo;
    float v1 = acc1[r] + bo;
    size_t yb = ((size_t)b * O + o) * M + m_base + row;
    Y[yb]      = v0;
    Y[yb + 16] = v1;
    float s1 = v0 + v1;
    float s2 = v0 * v0 + v1 * v1;
    for (int d = 1; d < 16; d <<= 1) {
      s1 += __shfl_xor(s1, d, 32);
      s2 += __shfl_xor(s2, d, 32);
    }
    if (row == 0) {
      atomicAdd(&stats[o], s1);
      atomicAdd(&stats[O + o], s2);
    }
  }
}

__global__ __launch_bounds__(256) void zero_buf(float* __restrict__ p, int n) {
  int i = blockIdx.x * 256 + threadIdx.x;
  if (i < n) p[i] = 0.f;
}

// per-channel BN scale/shift from accumulated sums (population variance)
__global__ __launch_bounds__(256) void bn_finalize(const float* __restrict__ stats,
                                                   const float* __restrict__ g,
                                                   const float* __restrict__ beta,
                                                   float* __restrict__ scsh,
                                                   int O, float cnt) {
  int o = threadIdx.x;
  if (o >= O) return;
  float m = stats[o] / cnt;
  float v = stats[O + o] / cnt - m * m;
  float sc = g[o] * rsqrtf(v + 1e-5f);
  scsh[o] = sc;
  scsh[O + o] = beta[o] - m * sc;
}

// dst = relu(scale * src + shift)
__global__ __launch_bounds__(256) void bn_apply(const float* __restrict__ src,
                                                float* __restrict__ dst,
                                                const float* __restrict__ scsh,
                                                int O, int M, size_t total) {
  size_t i = (size_t)blockIdx.x * 256 + threadIdx.x;
  if (i >= total) return;
  int o = (int)((i / (size_t)M) % O);
  float v = src[i] * scsh[o] + scsh[O + o];
  dst[i] = v > 0.f ? v : 0.f;
}

// max over K:  out[b,o,s] = max_k y[b,o,s*K+k]
__global__ __launch_bounds__(256) void maxpool_k(const float* __restrict__ y,
                                                 float* __restrict__ out,
                                                 int O, int S, int K) {
  size_t gid = (size_t)blockIdx.x * 256 + threadIdx.x;
  size_t tot = (size_t)BB * O * S;
  if (gid >= tot) return;
  size_t b = gid / ((size_t)O * S);
  size_t rem = gid % ((size_t)O * S);
  int o = (int)(rem / S), s = (int)(rem % S);
  const float* p = y + ((size_t)b * O + o) * ((size_t)S * K) + (size_t)s * K;
  float mx = p[0];
  for (int k = 1; k < K; ++k) mx = fmaxf(mx, p[k]);
  out[((size_t)b * O + o) * S + s] = mx;
}

// ---------------------------------------------------------------------------
// host orchestration
// ---------------------------------------------------------------------------
static inline unsigned gsz(size_t t) { return (unsigned)((t + 255) / 256); }

extern "C" void kernel_launch(void* const* d_in, const int* in_sizes, int n_in,
                              void* d_out, int out_size, void* d_ws, size_t ws_size,
                              hipStream_t stream) {
  (void)out_size; (void)ws_size;
  const float* input = (const float*)d_in[0];

  // --- resolve param pointers (handle both dict flattening orders) ---
  static const int NG = 6;  // canonical order: sa1, sa2, sa3, fp1, fp2, fp3
  static const int nl[NG] = {3, 2, 2, 2, 2, 3};
  static const int cin[NG][3]  = {{6, 64, 64}, {131, 128, 0}, {259, 256, 0},
                                  {384, 256, 0}, {384, 256, 0}, {134, 128, 128}};
  static const int cout[NG][3] = {{64, 64, 128}, {128, 256, 0}, {256, 128, 0},
                                  {256, 256, 0}, {256, 128, 0}, {128, 128, 128}};
  const int ordA[NG] = {0, 1, 2, 3, 4, 5};   // insertion order (sa first)
  const int ordB[NG] = {3, 4, 5, 0, 1, 2};   // alphabetical (fp first)
  auto matches = [&](const int* ord) -> bool {
    int idx = 1;
    for (int oi = 0; oi < NG; ++oi) {
      int g = ord[oi];
      for (int l = 0; l < nl[g]; ++l) {
        if (idx + 3 >= n_in) return false;
        if (in_sizes[idx]     != cout[g][l] * cin[g][l]) return false;
        if (in_sizes[idx + 1] != cout[g][l]) return false;
        if (in_sizes[idx + 2] != cout[g][l]) return false;
        if (in_sizes[idx + 3] != cout[g][l]) return false;
        idx += 4;
      }
    }
    return true;
  };
  const int* ord = matches(ordA) ? ordA : (matches(ordB) ? ordB : ordA);
  struct LayerPtr { const float *W, *b, *g, *be; int C, O; };
  LayerPtr L[NG][3];
  {
    int idx = 1;
    for (int oi = 0; oi < NG; ++oi) {
      int g = ord[oi];
      for (int l = 0; l < nl[g]; ++l) {
        if (idx + 3 < n_in)
          L[g][l] = {(const float*)d_in[idx], (const float*)d_in[idx + 1],
                     (const float*)d_in[idx + 2], (const float*)d_in[idx + 3],
                     cin[g][l], cout[g][l]};
        idx += 4;
      }
    }
  }

  // --- workspace layout (floats) ---
  float* base = (float*)d_ws;
  size_t off = 0;
  auto alloc = [&](size_t n) {
    float* p = base + off;
    off += (n + 63) & ~(size_t)63;
    return p;
  };
  float* xyz0  = alloc((size_t)BB * NN * 3);
  float* feat0 = alloc((size_t)BB * 6 * NN);
  float* l1xyz = alloc((size_t)BB * 512 * 3);
  float* l1pts = alloc((size_t)BB * 128 * 512);
  float* l2xyz = alloc((size_t)BB * 128 * 3);
  float* l2pts = alloc((size_t)BB * 256 * 128);
  float* l3pts = alloc((size_t)BB * 128);
  float* l2f   = alloc((size_t)BB * 256 * 128);
  float* l1f   = alloc((size_t)BB * 128 * 512);
  float* kw    = alloc((size_t)BB * NN * 3);
  float* stats = alloc(512);
  float* scsh  = alloc(512);
  float* wpad  = alloc((size_t)256 * 384);   // max O*Cpad
  int* fidx1 = (int*)alloc((size_t)BB * 512);
  int* fidx2 = (int*)alloc((size_t)BB * 128);
  int* ball1 = (int*)alloc((size_t)BB * 512 * 32);
  int* ball2 = (int*)alloc((size_t)BB * 128 * 64);
  int* kidx  = (int*)alloc((size_t)BB * NN * 3);
  float* actA = alloc((size_t)BB * 132 * 16384);  // fits all padded stages
  float* actB = alloc((size_t)BB * 132 * 16384);

  auto mlp_layer = [&](const LayerPtr& ly, const float* X, float* Y, int M,
                       float* dst) {
    int Cpad = (ly.C + 3) & ~3;
    pad_w<<<dim3(gsz((size_t)ly.O * Cpad)), dim3(256), 0, stream>>>(
        ly.W, wpad, ly.C, Cpad, ly.O);
    zero_buf<<<dim3(1), dim3(256), 0, stream>>>(stats, 2 * ly.O);
    dim3 gg((unsigned)(M / 128), (unsigned)(ly.O / 32), (unsigned)BB);
    gemm_bn_stats<<<gg, dim3(256), 0, stream>>>(X, wpad, ly.b, Y, stats,
                                                Cpad, ly.O, M);
    bn_finalize<<<dim3(1), dim3(256), 0, stream>>>(stats, ly.g, ly.be, scsh,
                                                   ly.O, (float)((size_t)BB * M));
    size_t tot = (size_t)BB * ly.O * M;
    bn_apply<<<dim3(gsz(tot)), dim3(256), 0, stream>>>(Y, dst ? dst : Y, scsh,
                                                       ly.O, M, tot);
  };

  // --- stage 0 ---
  split_input<<<dim3(gsz((size_t)BB * NN)), dim3(256), 0, stream>>>(input, xyz0, feat0);

  // --- SA1: 2048 -> 512 pts, r=0.2, K=32, MLP [64,64,128] (Cin 6 -> pad 8) ---
  fps_kernel<<<dim3(BB), dim3(256), 0, stream>>>(xyz0, fidx1, NN, 512);
  gather_new<<<dim3(gsz((size_t)BB * 512)), dim3(256), 0, stream>>>(xyz0, fidx1, l1xyz, NN, 512);
  query_ball<<<dim3(gsz((size_t)BB * 512)), dim3(256), 0, stream>>>(xyz0, l1xyz, ball1, NN, 512, 32, 0.04f);
  group_kernel<<<dim3(gsz((size_t)BB * 512 * 32)), dim3(256), 0, stream>>>(
      xyz0, l1xyz, feat0, ball1, actA, NN, 512, 32, /*Ctot=*/6, /*coff=*/3, /*Cp=*/3, /*Cpad=*/8);
  mlp_layer(L[0][0], actA, actB, 16384, nullptr);
  mlp_layer(L[0][1], actB, actA, 16384, nullptr);
  mlp_layer(L[0][2], actA, actB, 16384, nullptr);
  maxpool_k<<<dim3(gsz((size_t)BB * 128 * 512)), dim3(256), 0, stream>>>(actB, l1pts, 128, 512, 32);

  // --- SA2: 512 -> 128 pts, r=0.4, K=64, MLP [128,256] (Cin 131 -> pad 132) ---
  fps_kernel<<<dim3(BB), dim3(256), 0, stream>>>(l1xyz, fidx2, 512, 128);
  gather_new<<<dim3(gsz((size_t)BB * 128)), dim3(256), 0, stream>>>(l1xyz, fidx2, l2xyz, 512, 128);
  query_ball<<<dim3(gsz((size_t)BB * 128)), dim3(256), 0, stream>>>(l1xyz, l2xyz, ball2, 512, 128, 64, 0.16f);
  group_kernel<<<dim3(gsz((size_t)BB * 128 * 64)), dim3(256), 0, stream>>>(
      l1xyz, l2xyz, l1pts, ball2, actA, 512, 128, 64, /*Ctot=*/128, /*coff=*/0, /*Cp=*/128, /*Cpad=*/132);
  mlp_layer(L[1][0], actA, actB, 8192, nullptr);
  mlp_layer(L[1][1], actB, actA, 8192, nullptr);
  maxpool_k<<<dim3(gsz((size_t)BB * 256 * 128)), dim3(256), 0, stream>>>(actA, l2pts, 256, 128, 64);

  // --- SA3: global, MLP [256,128] (Cin 259 -> pad 260), max over 128 pts ---
  build_sa3<<<dim3(gsz((size_t)BB * 260 * 128)), dim3(256), 0, stream>>>(l2xyz, l2pts, actA);
  mlp_layer(L[2][0], actA, actB, 128, nullptr);
  mlp_layer(L[2][1], actB, actA, 128, nullptr);
  maxpool_k<<<dim3(gsz((size_t)BB * 128)), dim3(256), 0, stream>>>(actA, l3pts, 128, 1, 128);

  // --- FP1: broadcast l3 (C2=128) + l2_pts (256) -> MLP [256,256] -> l2f ---
  build_fp_input<<<dim3(gsz((size_t)BB * 384 * 128)), dim3(256), 0, stream>>>(
      l3pts, l2pts, nullptr, nullptr, actA, 128, 1, 256, 128, /*Cpad=*/384, /*mode=*/0);
  mlp_layer(L[3][0], actA, actB, 128, nullptr);
  mlp_layer(L[3][1], actB, actA, 128, l2f);

  // --- FP2: 3-NN (512 from 128); interp l2f (256) + l1_pts (128) -> [256,128] -> l1f ---
  knn3_kernel<<<dim3(gsz((size_t)BB * 512)), dim3(256), 0, stream>>>(l1xyz, l2xyz, kidx, kw, 512, 128);
  build_fp_input<<<dim3(gsz((size_t)BB * 384 * 512)), dim3(256), 0, stream>>>(
      l2f, l1pts, kidx, kw, actA, 256, 128, 128, 512, /*Cpad=*/384, /*mode=*/1);
  mlp_layer(L[4][0], actA, actB, 512, nullptr);
  mlp_layer(L[4][1], actB, actA, 512, l1f);

  // --- FP3: 3-NN (2048 from 512); interp l1f (128) + feat0 (6) -> [128,128,128] -> out ---
  knn3_kernel<<<dim3(gsz((size_t)BB * NN)), dim3(256), 0, stream>>>(xyz0, l1xyz, kidx, kw, NN, 512);
  build_fp_input<<<dim3(gsz((size_t)BB * 134 * NN)), dim3(256), 0, stream>>>(
      l1f, feat0, kidx, kw, actA, 128, 512, 6, NN, /*Cpad=*/136, /*mode=*/1);
  mlp_layer(L[5][0], actA, actB, NN, nullptr);
  mlp_layer(L[5][1], actB, actA, NN, nullptr);
  mlp_layer(L[5][2], actA, actB, NN, (float*)d_out);
}